// Net1_2525440770737
// MI455X (gfx1250) — compile-verified
//
#include <hip/hip_runtime.h>
#include <hip/hip_bf16.h>
#include <math.h>

typedef float v2f __attribute__((ext_vector_type(2)));
typedef float v8f __attribute__((ext_vector_type(8)));

#define N_NODES 59
#define CH      3
#define EDIM    9
#define NSLOT   64            // replicated global accumulators in d_ws
#define EPB     128           // edges per block-iteration (8 waves x 16)

#define LOG2E_F 1.4426950408889634f
#define LN2_F   0.6931471805599453f

// fast gate = sigmoid(f) * softplus(s) using raw v_exp/v_log/v_rcp
__device__ __forceinline__ float gate_f(float f, float s) {
    // sigmoid(f) = 1 / (1 + 2^(-f*log2e))
    float ef  = __builtin_amdgcn_exp2f(-f * LOG2E_F);
    float sig = __builtin_amdgcn_rcpf(1.0f + ef);
    // softplus(s) = max(s,0) + ln2 * log2(1 + 2^(-|s|*log2e))
    float es  = __builtin_amdgcn_exp2f(-fabsf(s) * LOG2E_F);
    float sp  = fmaxf(s, 0.0f) + LN2_F * __builtin_amdgcn_logf(1.0f + es);
    return sig * sp;
}

__global__ void zero_partial_kernel(float* partial, int n) {
    int i = blockIdx.x * blockDim.x + threadIdx.x;
    if (i < n) partial[i] = 0.0f;
}

// Main edge kernel: streams edge_attr, computes gate pre-activations with
// V_WMMA_F32_16X16X4_F32 (Z[16x16] x W2[16x16], K padded 15->16, N padded 6->16),
// applies sigmoid*softplus, segment-sums into LDS, spills to a replicated
// global accumulator slot.
__global__ __launch_bounds__(256)
void cgconv_edge_kernel(const float* __restrict__ x,        // [59*3]
                        const int*   __restrict__ eidx,     // [2*E] (src row, then dst row)
                        const float* __restrict__ ea,       // [E*9]
                        const float* __restrict__ Wf,       // [3*15]
                        const float* __restrict__ bf,       // [3]
                        const float* __restrict__ Ws,       // [3*15]
                        const float* __restrict__ bs,       // [3]
                        float*       __restrict__ partial,  // [NSLOT*177]
                        int E) {
    __shared__ float z_all[EPB][16];     // Z tile, K-major rows
    __shared__ float pre_all[EPB][6];    // WMMA outputs (f:0..2, s:3..5)
    __shared__ int   dst_all[EPB];
    __shared__ float x_s[192];           // node features
    __shared__ float wmat[16][16];       // W2: [k][n]
    __shared__ float acc_s[192];         // per-block segment sum (177 used)
    __shared__ float bf_s[4], bs_s[4];

    const int tid  = threadIdx.x;
    const int lane = tid & 31;
    const int wv   = tid >> 5;           // wave id 0..7

    // ---- one-time block setup ----
    if (tid < N_NODES * CH) x_s[tid] = x[tid];
    if (tid < 192) acc_s[tid] = 0.0f;
    if (tid < CH) { bf_s[tid] = bf[tid]; bs_s[tid] = bs[tid]; }
    {   // build padded W2[k][n]: n<3 -> Wf[n][k], 3<=n<6 -> Ws[n-3][k], else 0
        int k = tid >> 4, n = tid & 15;
        float v = 0.0f;
        if (k < 15) {
            if (n < 3)      v = Wf[n * 15 + k];
            else if (n < 6) v = Ws[(n - 3) * 15 + k];
        }
        wmat[k][n] = v;
    }
    __syncthreads();

    // loop-invariant B fragments (B is 4x16 per step; lane holds N=lane&15,
    // VGPR0/1 hold K = 2*kh / 2*kh+1 within the 4-wide K block, kh = lane>>4)
    v2f bfrag[4];
    {
        int n = lane & 15, kh = lane >> 4;
        #pragma unroll
        for (int kb = 0; kb < 4; ++kb) {
            bfrag[kb][0] = wmat[4 * kb + 2 * kh + 0][n];
            bfrag[kb][1] = wmat[4 * kb + 2 * kh + 1][n];
        }
    }

    for (int eBase = blockIdx.x * EPB; eBase < E; eBase += gridDim.x * EPB) {
        __syncthreads();   // protect z_all/dst_all from previous Phase C readers
        const int rem = min(EPB, E - eBase);

        // ---- Phase A: stage Z tile ----
        if (rem == EPB) {
            // 128 edges * 9 f32 = 1152 floats = 288 float4, base is 16B aligned
            const float4* p = (const float4*)(ea + (size_t)eBase * EDIM);
            for (int i = tid; i < (EPB * EDIM) / 4; i += 256) {
                float4 q = p[i];
                int lf = 4 * i;
                z_all[(lf + 0) / 9][6 + (lf + 0) % 9] = q.x;
                z_all[(lf + 1) / 9][6 + (lf + 1) % 9] = q.y;
                z_all[(lf + 2) / 9][6 + (lf + 2) % 9] = q.z;
                z_all[(lf + 3) / 9][6 + (lf + 3) % 9] = q.w;
            }
        } else {
            for (int g = tid; g < EPB * EDIM; g += 256) {
                float v = (g < rem * EDIM) ? ea[(size_t)eBase * EDIM + g] : 0.0f;
                z_all[g / 9][6 + g % 9] = v;
            }
        }
        if (tid < EPB) {
            int e = eBase + tid;
            int s = 0, d = 0;
            if (tid < rem) { s = eidx[e]; d = eidx[E + e]; }
            dst_all[tid] = d;
            #pragma unroll
            for (int c = 0; c < CH; ++c) {
                float vd = (tid < rem) ? x_s[d * CH + c] : 0.0f;
                float vs = (tid < rem) ? x_s[s * CH + c] : 0.0f;
                z_all[tid][c]     = vd;   // x_i (dst)
                z_all[tid][3 + c] = vs;   // x_j (src)
            }
            z_all[tid][15] = 0.0f;        // K padding
        }
        __syncthreads();

        // ---- Phase B: WMMA (each wave: 16 edges, 4 chained K-steps) ----
        {
            const int m16 = lane & 15, kh = lane >> 4;
            const float* zrow = &z_all[wv * 16 + m16][0];
            v8f acc = {};
            #pragma unroll
            for (int kb = 0; kb < 4; ++kb) {
                v2f a = *(const v2f*)(zrow + 4 * kb + 2 * kh);  // 8B aligned
                acc = __builtin_amdgcn_wmma_f32_16x16x4_f32(
                        false, a, false, bfrag[kb], (short)0, acc, false, false);
            }
            // D layout: VGPR r, lanes 0-15 -> (M=r, N=lane); 16-31 -> (M=8+r, N=lane-16)
            int n = lane & 15;
            if (n < 6) {
                #pragma unroll
                for (int r = 0; r < 8; ++r)
                    pre_all[wv * 16 + r + 8 * kh][n] = acc[r];
            }
        }
        __syncthreads();

        // ---- Phase C: activations + LDS segment-sum ----
        #pragma unroll
        for (int pass = 0; pass < 2; ++pass) {
            int p = tid + pass * 256;                 // 384 tasks = 128 edges x 3 ch
            if (p < EPB * CH) {
                int m = p / 3, c = p % 3;
                if (m < rem) {
                    float f = pre_all[m][c]     + bf_s[c];
                    float s = pre_all[m][c + 3] + bs_s[c];
                    float g = gate_f(f, s);
                    atomicAdd(&acc_s[dst_all[m] * CH + c], g);   // ds_add_f32
                }
            }
        }
    }

    // ---- epilogue: spill block sum into replicated global slot ----
    __syncthreads();
    if (tid < N_NODES * CH)
        atomicAdd(&partial[(blockIdx.x & (NSLOT - 1)) * (N_NODES * CH) + tid], acc_s[tid]);
}

// Final: h = relu(x + agg); out = dot(h, Wlin) + blin
__global__ __launch_bounds__(256)
void cgconv_finish_kernel(const float* __restrict__ x,
                          const float* __restrict__ Wlin,
                          const float* __restrict__ blin,
                          const float* __restrict__ partial,
                          float* __restrict__ out) {
    __shared__ float red[256];
    int t = threadIdx.x;
    float v = 0.0f;
    if (t < N_NODES * CH) {
        float s = 0.0f;
        #pragma unroll 4
        for (int k = 0; k < NSLOT; ++k) s += partial[k * (N_NODES * CH) + t];
        float h = fmaxf(x[t] + s, 0.0f);
        v = h * Wlin[t];
    }
    red[t] = v;
    __syncthreads();
    for (int off = 128; off > 0; off >>= 1) {
        if (t < off) red[t] += red[t + off];
        __syncthreads();
    }
    if (t == 0) out[0] = red[0] + blin[0];
}

extern "C" void kernel_launch(void* const* d_in, const int* in_sizes, int n_in,
                              void* d_out, int out_size, void* d_ws, size_t ws_size,
                              hipStream_t stream) {
    const float* x    = (const float*)d_in[0];
    const int*   eidx = (const int*)  d_in[1];
    const float* ea   = (const float*)d_in[2];
    // d_in[3] = batch (unused)
    const float* Wf   = (const float*)d_in[4];
    const float* bf   = (const float*)d_in[5];
    const float* Ws   = (const float*)d_in[6];
    const float* bs   = (const float*)d_in[7];
    const float* Wlin = (const float*)d_in[8];
    const float* blin = (const float*)d_in[9];
    float* out = (float*)d_out;
    float* partial = (float*)d_ws;          // NSLOT * 177 floats

    const int E = in_sizes[1] / 2;

    const int nPartial = NSLOT * N_NODES * CH;
    zero_partial_kernel<<<(nPartial + 255) / 256, 256, 0, stream>>>(partial, nPartial);

    int chunks = (E + EPB - 1) / EPB;
    int grid = chunks < 4096 ? (chunks > 0 ? chunks : 1) : 4096;
    cgconv_edge_kernel<<<grid, 256, 0, stream>>>(x, eidx, ea, Wf, bf, Ws, bs, partial, E);

    cgconv_finish_kernel<<<1, 256, 0, stream>>>(x, Wlin, blin, partial, out);
}